// EnhancedGNNModel_9105330668112
// MI455X (gfx1250) — compile-verified
//
#include <hip/hip_runtime.h>

#define NN    50000
#define NE    800000
#define FEAT  64
#define HID   128
#define NLAY  3

typedef __attribute__((ext_vector_type(2))) float v2f;
typedef __attribute__((ext_vector_type(8))) float v8f;

// ---------------------------------------------------------------------------
// utility: zero a float buffer (grid-stride)
__global__ __launch_bounds__(256) void zero_f32(float* p, int n) {
    int i = blockIdx.x * blockDim.x + threadIdx.x;
    int stride = gridDim.x * blockDim.x;
    for (; i < n; i += stride) p[i] = 0.0f;
}

// ---------------------------------------------------------------------------
// degree count (as float, used to fold segsum(h[dst]) = deg * h)
__global__ __launch_bounds__(256) void degree_kernel(const int* __restrict__ dst,
                                                     float* deg, int E) {
    int e = blockIdx.x * blockDim.x + threadIdx.x;
    if (e < E) atomicAdd(&deg[dst[e]], 1.0f);
}

// ---------------------------------------------------------------------------
// scatter-add: S[dst] += h[src].  One wave per edge; lane handles one float4
// (128 floats per row = 32 lanes * 4).  Gather + atomics stay L2-resident.
__global__ __launch_bounds__(256) void scatter_rows(const int* __restrict__ src,
                                                    const int* __restrict__ dst,
                                                    const float* __restrict__ h,
                                                    float* S, int E) {
    int t = blockIdx.x * blockDim.x + threadIdx.x;
    int e = t >> 5;
    int lane = t & 31;
    if (e >= E) return;
    int s = src[e];
    int d = dst[e];
    const float4* hv = (const float4*)(h + (size_t)s * HID);
    float4 v = hv[lane];
    float* o = S + (size_t)d * HID + (size_t)lane * 4;
    atomicAdd(o + 0, v.x);
    atomicAdd(o + 1, v.y);
    atomicAdd(o + 2, v.z);
    atomicAdd(o + 3, v.w);
}

// ---------------------------------------------------------------------------
// Fused dense GEMM:  out = act(A1@W1 + (rowscale2 .* A2)@W2 + bias) + residual
// A1/A2: [M, K] row-major, W1/W2: [K, HID] row-major, N fixed = HID = 128.
// Each wave computes MSUB=5 stacked 16x16 fp32 C tiles with
// V_WMMA_F32_16X16X4_F32, reusing every B fragment across the 5 tiles
// (5x less weight traffic from L2).  Block = 8 waves = 80 rows x 128 cols;
// 50000/80 = 625 blocks exactly, so EXEC is always all-ones (WMMA req).
// K is a template parameter so per-subtile A offsets constant-fold into the
// global_load immediate offset.
// In-place (out == A1 == residual) is safe: only this block touches its rows,
// and __syncthreads() separates all A reads from epilogue writes.
template <int K1, bool HAS2, bool RELU>
__global__ __launch_bounds__(256) void gemm_fused(
    const float* A1, const float* __restrict__ W1,
    const float* A2, const float* __restrict__ W2,
    const float* __restrict__ rowscale2,
    const float* __restrict__ bias,
    const float* residual,
    float* out)
{
    constexpr int K2   = HID;
    constexpr int MSUB = 5;                 // 5 x 16 = 80 rows per block
    const int lane = threadIdx.x & 31;
    const int wid  = threadIdx.x >> 5;      // 0..7 -> N tile
    const int m0   = blockIdx.x * (16 * MSUB);
    const int n0   = wid * 16;
    const int half = lane >> 4;             // ISA A layout: lanes 16-31 hold K+2
    const int l16  = lane & 15;

    v8f acc[MSUB];
    #pragma unroll
    for (int j = 0; j < MSUB; ++j) acc[j] = {};

    {   // acc += A1 @ W1
        const float* ap = A1 + (size_t)(m0 + l16) * K1 + 2 * half;
        const float* wp = W1 + (size_t)(2 * half) * HID + n0 + l16;
        #pragma unroll 4
        for (int k = 0; k < K1; k += 4) {
            v2f b;
            b.x = wp[(size_t)k * HID];                 // W[k+2h  ][col]
            b.y = wp[(size_t)(k + 1) * HID];           // W[k+2h+1][col]
            #pragma unroll
            for (int j = 0; j < MSUB; ++j) {
                v2f a = *(const v2f*)(ap + (size_t)(j * 16 * K1) + k);
                acc[j] = __builtin_amdgcn_wmma_f32_16x16x4_f32(
                    false, a, false, b, (short)0, acc[j], false, false);
            }
        }
    }
    if (HAS2) {  // acc += (rowscale2 .* A2) @ W2
        float s2[MSUB];
        #pragma unroll
        for (int j = 0; j < MSUB; ++j)
            s2[j] = rowscale2 ? rowscale2[m0 + j * 16 + l16] : 1.0f;
        const float* ap = A2 + (size_t)(m0 + l16) * K2 + 2 * half;
        const float* wp = W2 + (size_t)(2 * half) * HID + n0 + l16;
        #pragma unroll 4
        for (int k = 0; k < K2; k += 4) {
            v2f b;
            b.x = wp[(size_t)k * HID];
            b.y = wp[(size_t)(k + 1) * HID];
            #pragma unroll
            for (int j = 0; j < MSUB; ++j) {
                v2f a = *(const v2f*)(ap + (size_t)(j * 16 * K2) + k);
                a.x *= s2[j];
                a.y *= s2[j];
                acc[j] = __builtin_amdgcn_wmma_f32_16x16x4_f32(
                    false, a, false, b, (short)0, acc[j], false, false);
            }
        }
    }

    __syncthreads();  // all A reads done before any in-place epilogue write

    // C layout (ISA 7.12.2): VGPR i -> row 16*tile + half*8 + i, col n0 + l16
    const int col = n0 + l16;
    const float bv = bias[col];
    #pragma unroll
    for (int j = 0; j < MSUB; ++j) {
        #pragma unroll
        for (int i = 0; i < 8; ++i) {
            int r = m0 + j * 16 + half * 8 + i;
            float c = acc[j][i] + bv;
            if (RELU) c = fmaxf(c, 0.0f);
            if (residual) c += residual[(size_t)r * HID + col];
            out[(size_t)r * HID + col] = c;
        }
    }
}

// ---------------------------------------------------------------------------
// column-sum partial reduction for the mean pool (atomics into gsum[128])
__global__ __launch_bounds__(128) void colsum_kernel(const float* __restrict__ h,
                                                     float* gsum, int M, int rpb) {
    int c = threadIdx.x;
    int r0 = blockIdx.x * rpb;
    int r1 = r0 + rpb; if (r1 > M) r1 = M;
    float s = 0.0f;
    for (int r = r0; r < r1; ++r) s += h[(size_t)r * HID + c];
    atomicAdd(&gsum[c], s);
}

// ---------------------------------------------------------------------------
// output MLP head: [1,128]->relu[64]->relu[32]->[1]
__global__ __launch_bounds__(128) void head_kernel(
    const float* __restrict__ gsum,
    const float* __restrict__ W1, const float* __restrict__ b1,
    const float* __restrict__ W2, const float* __restrict__ b2,
    const float* __restrict__ W3, const float* __restrict__ b3,
    float* out, float invM)
{
    __shared__ float g[HID];
    __shared__ float z1[64];
    __shared__ float z2[32];
    int t = threadIdx.x;
    g[t] = gsum[t] * invM;
    __syncthreads();
    if (t < 64) {
        float s = b1[t];
        for (int k = 0; k < HID; ++k) s += g[k] * W1[k * 64 + t];
        z1[t] = fmaxf(s, 0.0f);
    }
    __syncthreads();
    if (t < 32) {
        float s = b2[t];
        for (int k = 0; k < 64; ++k) s += z1[k] * W2[k * 32 + t];
        z2[t] = fmaxf(s, 0.0f);
    }
    __syncthreads();
    if (t == 0) {
        float s = b3[0];
        for (int k = 0; k < 32; ++k) s += z2[k] * W3[k];
        out[0] = s;
    }
}

// ---------------------------------------------------------------------------
extern "C" void kernel_launch(void* const* d_in, const int* in_sizes, int n_in,
                              void* d_out, int out_size, void* d_ws, size_t ws_size,
                              hipStream_t stream) {
    const float* atom   = (const float*)d_in[0];   // [NN, FEAT]
    const int*   ei     = (const int*)  d_in[1];   // [2, NE]
    const float* emb_W  = (const float*)d_in[2];   // [FEAT, HID]
    const float* emb_b  = (const float*)d_in[3];   // [HID]
    const float* msg_W  = (const float*)d_in[4];   // [L, 2H, H]
    const float* msg_b  = (const float*)d_in[5];   // [L, H]
    const float* upd_W  = (const float*)d_in[6];   // [L, 2H, H]
    const float* upd_b  = (const float*)d_in[7];   // [L, H]
    const float* out_W1 = (const float*)d_in[8];
    const float* out_b1 = (const float*)d_in[9];
    const float* out_W2 = (const float*)d_in[10];
    const float* out_b2 = (const float*)d_in[11];
    const float* out_W3 = (const float*)d_in[12];
    const float* out_b3 = (const float*)d_in[13];

    const int* src = ei;            // edge_indices[0]
    const int* dst = ei + NE;       // edge_indices[1]

    // workspace partition (~77 MB, all L2-resident on MI455X)
    float* h    = (float*)d_ws;                        // NN*HID
    float* S    = h   + (size_t)NN * HID;              // NN*HID
    float* agg  = S   + (size_t)NN * HID;              // NN*HID
    float* deg  = agg + (size_t)NN * HID;              // NN
    float* gsum = deg + NN;                            // HID

    const int MB = NN / 80;  // 625 row blocks, exact

    // degree (recomputed every call -> deterministic) + zero mean accumulator
    zero_f32<<<512, 256, 0, stream>>>(deg, NN + HID);
    degree_kernel<<<(NE + 255) / 256, 256, 0, stream>>>(dst, deg, NE);

    // atom embedding: h = atom @ emb_W + emb_b
    gemm_fused<FEAT, false, false><<<MB, 256, 0, stream>>>(
        atom, emb_W, nullptr, nullptr, nullptr, emb_b, nullptr, h);

    for (int l = 0; l < NLAY; ++l) {
        const float* Wm = msg_W + (size_t)l * 2 * HID * HID;
        const float* Wu = upd_W + (size_t)l * 2 * HID * HID;

        zero_f32<<<2048, 256, 0, stream>>>(S, NN * HID);
        scatter_rows<<<(NE * 32) / 256, 256, 0, stream>>>(src, dst, h, S, NE);

        // agg = S @ Wm_top + (deg .* h) @ Wm_bot + msg_b[l]
        gemm_fused<HID, true, false><<<MB, 256, 0, stream>>>(
            S, Wm, h, Wm + HID * HID, deg,
            msg_b + (size_t)l * HID, nullptr, agg);

        // h = relu(h @ Wu_top + agg @ Wu_bot + upd_b[l]) + h   (in-place)
        gemm_fused<HID, true, true><<<MB, 256, 0, stream>>>(
            h, Wu, agg, Wu + HID * HID, nullptr,
            upd_b + (size_t)l * HID, h, h);
    }

    // mean pool + output MLP
    colsum_kernel<<<200, 128, 0, stream>>>(h, gsum, NN, 250);
    head_kernel<<<1, 128, 0, stream>>>(gsum, out_W1, out_b1, out_W2, out_b2,
                                       out_W3, out_b3, (float*)d_out,
                                       1.0f / (float)NN);
}